// VariationalGCNEncoder_46093589021377
// MI455X (gfx1250) — compile-verified
//
#include <hip/hip_runtime.h>

typedef float v2f __attribute__((ext_vector_type(2)));
typedef float v8f __attribute__((ext_vector_type(8)));

// ---------------------------------------------------------------------------
// init: deg = 1.0 (self loop), zero layer-1 scatter accumulator, zero d_out
// (d_out holds mu||logstd = N*128 floats total)
// ---------------------------------------------------------------------------
__global__ __launch_bounds__(256) void gcn_init_kernel(float* __restrict__ deg,
                                                       float* __restrict__ agg,
                                                       float* __restrict__ out,
                                                       int N) {
  const long total = (long)N * 128;
  for (long t = (long)blockIdx.x * blockDim.x + threadIdx.x; t < total;
       t += (long)gridDim.x * blockDim.x) {
    agg[t] = 0.0f;
    out[t] = 0.0f;
    if (t < N) deg[t] = 1.0f;
  }
}

// ---------------------------------------------------------------------------
// pack W_mu (128x64) and W_ls (128x64) into Wcat (128x128) = [W_mu | W_ls]
// ---------------------------------------------------------------------------
__global__ __launch_bounds__(256) void gcn_pack_w_kernel(const float* __restrict__ Wmu,
                                                         const float* __restrict__ Wls,
                                                         float* __restrict__ Wcat) {
  const int t = blockIdx.x * blockDim.x + threadIdx.x;
  if (t < 128 * 64) {
    const int k = t >> 6, j = t & 63;
    Wcat[k * 128 + j] = Wmu[t];
    Wcat[k * 128 + 64 + j] = Wls[t];
  }
}

// ---------------------------------------------------------------------------
// degree: deg[dst] += 1 per edge (atomics land in L2; 400KB array stays hot)
// ---------------------------------------------------------------------------
__global__ __launch_bounds__(256) void gcn_degree_kernel(const int* __restrict__ ei,
                                                         float* __restrict__ deg,
                                                         int E) {
  const int e = blockIdx.x * blockDim.x + threadIdx.x;
  if (e < E) atomicAdd(&deg[ei[(long)E + e]], 1.0f);
}

__global__ __launch_bounds__(256) void gcn_rsqrt_kernel(float* __restrict__ deg, int N) {
  const int i = blockIdx.x * blockDim.x + threadIdx.x;
  if (i < N) deg[i] = rsqrtf(deg[i]);  // deg >= 1 always
}

// ---------------------------------------------------------------------------
// C[M x 128] = A[M x 128] @ B[128 x 128] using V_WMMA_F32_16X16X4_F32.
// 256 threads = 8 waves; wave w owns rows [m0, m0+16). B is staged in 64KB LDS
// with an XOR-16 swizzle on ((row>>1)&1) so the two 16-lane halves of a wave
// (which read rows k/k+1 vs k+2/k+3) hit disjoint bank groups.
//
// Fragment layouts (ISA 7.12.2, 32-bit, wave32):
//   A 16x4: lane<16 -> {K=kk+0, kk+1}; lane>=16 -> {K=kk+2, kk+3}; M = lane&15
//   B 4x16: same K split; N = lane&15
//   C 16x16: acc[r] -> row m0 + r + 8*(lane>>4), col = lane&15 (+16*nt tile)
// ---------------------------------------------------------------------------
__global__ __launch_bounds__(256) void gcn_gemm_wmma_f32(const float* __restrict__ A,
                                                         const float* __restrict__ Bg,
                                                         float* __restrict__ C,
                                                         int M) {
  __shared__ float Bs[128 * 128];

  for (int idx = threadIdx.x; idx < 128 * 32; idx += 256) {
    const int r = idx >> 5;
    const int c4 = (idx & 31) << 2;
    const int sw = ((r >> 1) & 1) << 4;
    const float4 v = *(const float4*)(Bg + r * 128 + c4);
    *(float4*)(&Bs[r * 128 + (c4 ^ sw)]) = v;
  }
  __syncthreads();

  const int lane = threadIdx.x & 31;
  const int wave = threadIdx.x >> 5;
  const int m0 = (blockIdx.x * 8 + wave) * 16;
  if (m0 >= M) return;

  const int lm = lane & 15;
  const int half = lane >> 4;
  const int koff = half * 2;

  int arow = m0 + lm;
  if (arow > M - 1) arow = M - 1;  // clamp loads; stores are guarded
  const float* Ap = A + (size_t)arow * 128 + koff;

  v8f acc[8];
#pragma unroll
  for (int nt = 0; nt < 8; ++nt) {
    v8f z = {0.f, 0.f, 0.f, 0.f, 0.f, 0.f, 0.f, 0.f};
    acc[nt] = z;
  }

  for (int kk = 0; kk < 128; kk += 4) {
    const float2 av = *(const float2*)(Ap + kk);
    v2f a;
    a.x = av.x;
    a.y = av.y;
    const int kb = kk + koff;
    const int sw = ((kb >> 1) & 1) << 4;
    const float* Bp = &Bs[kb * 128];
#pragma unroll
    for (int nt = 0; nt < 8; ++nt) {
      const int c = ((nt * 16) ^ sw) + lm;
      v2f b;
      b.x = Bp[c];
      b.y = Bp[128 + c];
      acc[nt] = __builtin_amdgcn_wmma_f32_16x16x4_f32(
          false, a, false, b, (short)0, acc[nt], false, false);
    }
  }

#pragma unroll
  for (int nt = 0; nt < 8; ++nt) {
#pragma unroll
    for (int r = 0; r < 8; ++r) {
      const int crow = m0 + r + half * 8;
      if (crow < M) C[(size_t)crow * 128 + nt * 16 + lm] = acc[nt][r];
    }
  }
}

// ---------------------------------------------------------------------------
// Edge scatter: 32 lanes per edge, each lane handles 4 consecutive features.
// agg[dst] += H[src] * dinv[src]*dinv[dst].  SPLIT=true routes cols [0,64)
// to out0 (mu) and [64,128) to out1 (logstd), each N x 64.
// Targets fit in L2 (51.2MB), so the atomic RMW traffic stays on-chip.
// ---------------------------------------------------------------------------
template <bool SPLIT>
__global__ __launch_bounds__(256) void gcn_scatter_kernel(const float* __restrict__ H,
                                                          const int* __restrict__ ei,
                                                          const float* __restrict__ dinv,
                                                          float* __restrict__ out0,
                                                          float* __restrict__ out1,
                                                          int E) {
  const long gid = (long)blockIdx.x * blockDim.x + threadIdx.x;
  const long edge = gid >> 5;
  const int lane = (int)(gid & 31);
  if (edge >= E) return;

  const int s = ei[edge];
  const int d = ei[(long)E + edge];
  const float nrm = dinv[s] * dinv[d];
  const float4 h = *(const float4*)(H + (size_t)s * 128 + lane * 4);

  float* base;
  if (SPLIT) {
    base = (lane < 16) ? (out0 + (size_t)d * 64 + lane * 4)
                       : (out1 + (size_t)d * 64 + (lane - 16) * 4);
  } else {
    base = out0 + (size_t)d * 128 + lane * 4;
  }
  atomicAdd(base + 0, h.x * nrm);
  atomicAdd(base + 1, h.y * nrm);
  atomicAdd(base + 2, h.z * nrm);
  atomicAdd(base + 3, h.w * nrm);
}

// ---------------------------------------------------------------------------
// Layer 1 epilogue: H = relu(agg + H*dinv^2 + b1), written in place over H.
// ---------------------------------------------------------------------------
__global__ __launch_bounds__(256) void gcn_finalize1_kernel(float* __restrict__ H,
                                                            const float* __restrict__ agg,
                                                            const float* __restrict__ dinv,
                                                            const float* __restrict__ b1,
                                                            int N) {
  const long t = (long)blockIdx.x * blockDim.x + threadIdx.x;
  if (t >= (long)N * 128) return;
  const int i = (int)(t >> 7);
  const int c = (int)(t & 127);
  const float di = dinv[i];
  const float v = agg[t] + H[t] * di * di + b1[c];
  H[t] = v > 0.0f ? v : 0.0f;
}

// ---------------------------------------------------------------------------
// Layer 2 epilogue: out already holds the scatter sums; add self-loop + bias.
// Hcat row i = [h@W_mu (64) | h@W_ls (64)].
// ---------------------------------------------------------------------------
__global__ __launch_bounds__(256) void gcn_finalize2_kernel(const float* __restrict__ Hcat,
                                                            const float* __restrict__ dinv,
                                                            const float* __restrict__ bmu,
                                                            const float* __restrict__ bls,
                                                            float* __restrict__ out,
                                                            int N) {
  const long t = (long)blockIdx.x * blockDim.x + threadIdx.x;
  if (t >= (long)N * 64) return;
  const int i = (int)(t >> 6);
  const int c = (int)(t & 63);
  const float di = dinv[i];
  const float d2 = di * di;
  const float* hr = Hcat + (size_t)i * 128;
  out[t] += hr[c] * d2 + bmu[c];
  out[(size_t)N * 64 + t] += hr[64 + c] * d2 + bls[c];
}

extern "C" void kernel_launch(void* const* d_in, const int* in_sizes, int n_in,
                              void* d_out, int out_size, void* d_ws, size_t ws_size,
                              hipStream_t stream) {
  (void)n_in; (void)out_size; (void)ws_size;
  const float* x   = (const float*)d_in[0];
  const int*   ei  = (const int*)d_in[1];   // [2, E] flattened: src then dst
  const float* W1  = (const float*)d_in[2];
  const float* b1  = (const float*)d_in[3];
  const float* Wmu = (const float*)d_in[4];
  const float* bmu = (const float*)d_in[5];
  const float* Wls = (const float*)d_in[6];
  const float* bls = (const float*)d_in[7];
  float* out = (float*)d_out;

  const int N = in_sizes[0] / 128;
  const int E = in_sizes[1] / 2;

  // Workspace carve: dinv(N) | H1(N*128) | AGG(N*128, reused as Hcat) | Wcat
  char* ws = (char*)d_ws;
  size_t off = 0;
  auto carve = [&](size_t bytes) -> float* {
    float* p = (float*)(ws + off);
    off = (off + bytes + 255) & ~(size_t)255;
    return p;
  };
  float* deg  = carve((size_t)N * 4);            // becomes dinv in place
  float* H1   = carve((size_t)N * 128 * 4);      // X@W1, then relu'd hidden
  float* AGG  = carve((size_t)N * 128 * 4);      // layer-1 accum, then Hcat
  float* Wcat = carve((size_t)128 * 128 * 4);

  const int T = 256;
  const long n128 = (long)N * 128;
  const long n64  = (long)N * 64;
  const long ethreads = (long)E * 32;

  gcn_init_kernel<<<(int)((n128 + T - 1) / T), T, 0, stream>>>(deg, AGG, out, N);
  gcn_pack_w_kernel<<<(128 * 64 + T - 1) / T, T, 0, stream>>>(Wmu, Wls, Wcat);
  gcn_degree_kernel<<<(E + T - 1) / T, T, 0, stream>>>(ei, deg, E);
  gcn_rsqrt_kernel<<<(N + T - 1) / T, T, 0, stream>>>(deg, N);

  // Layer 1: H1 = X @ W1
  gcn_gemm_wmma_f32<<<(N + 127) / 128, T, 0, stream>>>(x, W1, H1, N);
  gcn_scatter_kernel<false><<<(int)((ethreads + T - 1) / T), T, 0, stream>>>(
      H1, ei, deg, AGG, nullptr, E);
  gcn_finalize1_kernel<<<(int)((n128 + T - 1) / T), T, 0, stream>>>(H1, AGG, deg, b1, N);

  // Layer 2 (mu and logstd fused): Hcat = H1 @ [W_mu | W_ls]
  gcn_gemm_wmma_f32<<<(N + 127) / 128, T, 0, stream>>>(H1, Wcat, AGG, N);
  gcn_scatter_kernel<true><<<(int)((ethreads + T - 1) / T), T, 0, stream>>>(
      AGG, ei, deg, out, out + (size_t)N * 64, E);
  gcn_finalize2_kernel<<<(int)((n64 + T - 1) / T), T, 0, stream>>>(
      AGG, deg, bmu, bls, out, N);
}